// YOLOLoss_53472342835728
// MI455X (gfx1250) — compile-verified
//
#include <hip/hip_runtime.h>
#include <hip/hip_bf16.h>
#include <stdint.h>

// ---------------------------------------------------------------------------
// YOLO loss for MI455X (gfx1250, wave32)
//   B=16 images, N=16384 anchors, M=128 gt boxes, C=80 classes
//   Kernel 1: per-anchor best-GIoU matching (gt staged in LDS via
//             global_load_async_to_lds_*), masked box + focal cls loss,
//             deterministic per-block partial sums into d_ws.
//   Kernel 2: single-wave reduction of the 1024 partials per quantity using
//             V_WMMA_F32_16X16X4_F32 with an all-ones B matrix (row-sum
//             trick), then writes the final scalar loss.
// ---------------------------------------------------------------------------

#define NUM_CLASSES 80
#define BIMG   16
#define NANCH  16384
#define MGT    128
#define ALPHA_F 0.25f
#define IOU_THR 0.3f
#define BLOCK   256
#define NBLK    ((NANCH / BLOCK) * BIMG)   // 1024 partial-sum slots

typedef __attribute__((ext_vector_type(2))) float v2f;
typedef __attribute__((ext_vector_type(8))) float v8f;

__device__ __forceinline__ float focal_term(float x, bool ispos) {
  // stable BCE-with-logits: ce = max(x,0) - x*t + log1p(exp(-|x|))
  float ax = fabsf(x);
  float e  = __expf(-ax);                     // v_exp_f32
  float lg = __logf(1.0f + e);                // v_log_f32 (log1p, e in (0,1])
  float t  = ispos ? 1.0f : 0.0f;
  float ce = fmaxf(x, 0.0f) - x * t + lg;
  float s  = (x >= 0.0f) ? (1.0f / (1.0f + e)) : (e / (1.0f + e)); // sigmoid
  float pt = ispos ? s : (1.0f - s);
  float at = ispos ? ALPHA_F : (1.0f - ALPHA_F);
  float om = 1.0f - pt;
  return at * om * om * ce;                   // gamma == 2 -> square
}

__global__ __launch_bounds__(BLOCK) void yolo_match_kernel(
    const float* __restrict__ box_preds,   // [B,N,4]
    const float* __restrict__ cls_preds,   // [B,N,C]
    const float* __restrict__ gt_boxes,    // [B,M,4]
    const int*   __restrict__ gt_labels,   // [B,M]
    float* __restrict__ partials)          // [3*NBLK] box / cls / npos
{
  __shared__ float4 sGT[MGT];
  __shared__ int    sLBL[MGT];
  __shared__ float  sred[3][BLOCK / 32];

  const int tid = threadIdx.x;
  const int b   = blockIdx.y;
  const int n   = blockIdx.x * BLOCK + tid;

  // ---- CDNA5 async global->LDS staging of gt boxes + labels --------------
  // Lanes 0..127 pull one float4 box each (b128); lanes 128..255 pull one
  // int32 label each (b32). Low 32 bits of the flat shared address are the
  // LDS byte offset (ISA aperture rule).
  if (tid < MGT) {
    unsigned lds = (unsigned)(uintptr_t)&sGT[tid];
    unsigned long long ga =
        (unsigned long long)(uintptr_t)(gt_boxes + ((size_t)b * MGT + tid) * 4);
    asm volatile("global_load_async_to_lds_b128 %0, %1, off"
                 :: "v"(lds), "v"(ga) : "memory");
  } else {
    int i = tid - MGT;
    unsigned lds = (unsigned)(uintptr_t)&sLBL[i];
    unsigned long long ga =
        (unsigned long long)(uintptr_t)(gt_labels + (size_t)b * MGT + i);
    asm volatile("global_load_async_to_lds_b32 %0, %1, off"
                 :: "v"(lds), "v"(ga) : "memory");
  }
  asm volatile("s_wait_asynccnt 0" ::: "memory");
  __syncthreads();

  // ---- per-anchor best-GIoU over the 128 LDS-resident gt boxes -----------
  const float4 pb =
      reinterpret_cast<const float4*>(box_preds)[(size_t)b * NANCH + n];
  const float px1 = pb.x, py1 = pb.y, px2 = pb.z, py2 = pb.w;
  const float areap = (px2 - px1) * (py2 - py1);

  float best = -3.0e38f;
  int   bidx = 0;
#pragma unroll 4
  for (int m = 0; m < MGT; ++m) {
    float4 g = sGT[m];                               // ds_load_b128
    float iw = fminf(px2, g.z) - fmaxf(px1, g.x);
    float ih = fminf(py2, g.w) - fmaxf(py1, g.y);
    iw = fmaxf(iw, 0.0f);
    ih = fmaxf(ih, 0.0f);
    float inter = iw * ih;
    float areag = (g.z - g.x) * (g.w - g.y);
    float uni   = areap + areag - inter;
    float iou   = inter / uni;
    float ew = fmaxf(px2, g.z) - fminf(px1, g.x);
    float eh = fmaxf(py2, g.w) - fminf(py1, g.y);
    float areae = fmaxf(ew, 0.0f) * fmaxf(eh, 0.0f);
    float giou  = iou - (areae - uni) / areae;
    if (giou > best) { best = giou; bidx = m; }      // first-max like argmax
  }

  // ---- losses (positives only; elementwise giou(pred, matched) == best) --
  float box_l = 0.0f, cls_l = 0.0f, npos = 0.0f;
  if (best > IOU_THR) {
    box_l = 1.0f - best;
    npos  = 1.0f;
    const int lbl = sLBL[bidx];
    const float4* crow = reinterpret_cast<const float4*>(
        cls_preds + ((size_t)b * NANCH + n) * NUM_CLASSES);
#pragma unroll
    for (int q = 0; q < NUM_CLASSES / 4; ++q) {
      float4 xv = crow[q];
      int c = q * 4;
      cls_l += focal_term(xv.x, c + 0 == lbl);
      cls_l += focal_term(xv.y, c + 1 == lbl);
      cls_l += focal_term(xv.z, c + 2 == lbl);
      cls_l += focal_term(xv.w, c + 3 == lbl);
    }
  }

  // ---- deterministic block reduction (wave32 shuffles, then LDS) ---------
#pragma unroll
  for (int o = 16; o > 0; o >>= 1) {
    box_l += __shfl_down(box_l, o, 32);
    cls_l += __shfl_down(cls_l, o, 32);
    npos  += __shfl_down(npos,  o, 32);
  }
  const int lane = tid & 31, wid = tid >> 5;
  if (lane == 0) { sred[0][wid] = box_l; sred[1][wid] = cls_l; sred[2][wid] = npos; }
  __syncthreads();
  if (tid == 0) {
    float tb = 0.0f, tc = 0.0f, tn = 0.0f;
#pragma unroll
    for (int w = 0; w < BLOCK / 32; ++w) {
      tb += sred[0][w]; tc += sred[1][w]; tn += sred[2][w];
    }
    const int pidx = blockIdx.y * gridDim.x + blockIdx.x;
    partials[pidx]            = tb;
    partials[NBLK + pidx]     = tc;
    partials[2 * NBLK + pidx] = tn;
  }
}

// ---------------------------------------------------------------------------
// Final reduction: one wave32; sum 1024 fp32 partials per quantity with
// V_WMMA_F32_16X16X4_F32 against an all-ones B matrix (D = A*ones + C
// accumulates row-sums; 64 values per WMMA). EXEC is all-ones (32 threads,
// no divergence before the WMMAs) as the ISA requires.
// ---------------------------------------------------------------------------
__global__ __launch_bounds__(32) void yolo_finalize_kernel(
    const float* __restrict__ partials, float* __restrict__ out)
{
  const int lane = threadIdx.x;
  float tot[3];
#if __has_builtin(__builtin_amdgcn_wmma_f32_16x16x4_f32)
  const v2f bones = {1.0f, 1.0f};
#pragma unroll
  for (int a = 0; a < 3; ++a) {
    const float* v = partials + a * NBLK;
    v8f acc = {};
    for (int ch = 0; ch < NBLK / 64; ++ch) {
      v2f av = {v[ch * 64 + lane], v[ch * 64 + 32 + lane]};
      acc = __builtin_amdgcn_wmma_f32_16x16x4_f32(
          false, av, false, bones, (short)0, acc, false, false);
    }
    // C/D layout: lane j holds rows 0..7 (j<16) / rows 8..15 (j>=16) of one
    // column; column total = my 8 regs + partner lane (j ^ 16).
    float s = ((acc[0] + acc[1]) + (acc[2] + acc[3])) +
              ((acc[4] + acc[5]) + (acc[6] + acc[7]));
    s += __shfl_xor(s, 16, 32);
    tot[a] = s;
  }
#else
#pragma unroll
  for (int a = 0; a < 3; ++a) {
    float s = 0.0f;
    for (int i = lane; i < NBLK; i += 32) s += partials[a * NBLK + i];
    for (int o = 16; o > 0; o >>= 1) s += __shfl_xor(s, o, 32);
    tot[a] = s;
  }
#endif
  if (lane == 0)
    out[0] = (5.0f * tot[0] + tot[1]) / fmaxf(tot[2], 1.0f);
}

extern "C" void kernel_launch(void* const* d_in, const int* in_sizes, int n_in,
                              void* d_out, int out_size, void* d_ws, size_t ws_size,
                              hipStream_t stream) {
  (void)in_sizes; (void)n_in; (void)out_size; (void)ws_size;
  const float* box_preds = (const float*)d_in[0];   // [16,16384,4] f32
  const float* cls_preds = (const float*)d_in[1];   // [16,16384,80] f32
  const float* gt_boxes  = (const float*)d_in[2];   // [16,128,4] f32
  const int*   gt_labels = (const int*)d_in[3];     // [16,128] i32
  float* out      = (float*)d_out;                  // scalar f32 loss
  float* partials = (float*)d_ws;                   // needs 3*1024*4 = 12 KB

  dim3 grid(NANCH / BLOCK, BIMG);                   // 1024 blocks total
  yolo_match_kernel<<<grid, BLOCK, 0, stream>>>(
      box_preds, cls_preds, gt_boxes, gt_labels, partials);
  yolo_finalize_kernel<<<1, 32, 0, stream>>>(partials, out);
}